// LLClusterCoordinates_49598282334780
// MI455X (gfx1250) — compile-verified
//
#include <hip/hip_runtime.h>
#include <hip/hip_bf16.h>
#include <math.h>

#define KMAX 200
#define KPAD 208            // padded to 13 tiles of 16
#define KTILES (KPAD / 16)
#define MAXEV 4

// workspace layout (in floats)
#define STAT_OFF 0                      // [MAXEV][KMAX][4] = {sx,sy,sz,cnt}
#define STAT_SIZE (MAXEV * KMAX * 4)    // 3200
#define AV4_OFF  (STAT_OFF + STAT_SIZE) // [MAXEV][KPAD] float4 = (-2ax,-2ay,-2az,|a|^2)
#define AV4_SIZE (MAXEV * KPAD * 4)     // 3328
#define IR_OFF   (AV4_OFF + AV4_SIZE)   // invRep [MAXEV][KPAD]
#define IR_SIZE  (MAXEV * KPAD)
#define ID_OFF   (IR_OFF + IR_SIZE)     // invDist [MAXEV][KPAD]
#define ID_SIZE  (MAXEV * KPAD)
#define AF_OFF   (ID_OFF + ID_SIZE)     // pre-swizzled A fragments [MAXEV][KTILES][32] float2
#define AF_SIZE  (MAXEV * KTILES * 32 * 2)

typedef __attribute__((ext_vector_type(2))) float v2f;
typedef __attribute__((ext_vector_type(8))) float v8f;

#if defined(__gfx1250__) && __has_builtin(__builtin_amdgcn_wmma_f32_16x16x4_f32)
#define USE_WMMA 1
#else
#define USE_WMMA 0
#endif

__device__ __forceinline__ float clamp01(float x) {
#if defined(__HIP_DEVICE_COMPILE__)
    // med3(x,0,1) is the AMDGPU clamp pattern: folds into the output clamp
    // modifier of the defining instruction (here v_exp_f32 ... clamp).
    return __builtin_amdgcn_fmed3f(x, 0.0f, 1.0f);
#else
    return x < 0.0f ? 0.0f : (x > 1.0f ? 1.0f : x);
#endif
}

__device__ __forceinline__ float clamp_pos(float x) {
#if defined(__HIP_DEVICE_COMPILE__)
    return __builtin_amdgcn_fmed3f(x, 0.0f, __builtin_inff());
#else
    return x > 0.0f ? x : 0.0f;
#endif
}

__global__ void ll_zero_kernel(float* ws, float* out) {
    int i = blockIdx.x * blockDim.x + threadIdx.x;
    if (i < STAT_SIZE) ws[STAT_OFF + i] = 0.0f;
    if (i == 0) out[0] = 0.0f;
}

// Accumulate per-cluster sums and counts via LDS staging, then global atomics.
__global__ void ll_accum_kernel(const float* __restrict__ coords,
                                const int* __restrict__ tidx,
                                const long long* __restrict__ rs,
                                float* ws, int nPoints, int nEvents) {
    __shared__ float ls[MAXEV * KMAX * 4];
    for (int i = threadIdx.x; i < MAXEV * KMAX * 4; i += blockDim.x) ls[i] = 0.0f;
    __syncthreads();
    for (int p = blockIdx.x * blockDim.x + threadIdx.x; p < nPoints;
         p += gridDim.x * blockDim.x) {
        int t = tidx[p];
        if (t >= 0 && t < KMAX) {
            int e = 0;
            for (int i = 0; i < nEvents; ++i)
                if ((long long)p >= rs[i]) e = i;
            if (e < MAXEV) {
                float x = coords[3 * p + 0];
                float y = coords[3 * p + 1];
                float z = coords[3 * p + 2];
                float* base = &ls[(e * KMAX + t) * 4];
                atomicAdd(base + 0, x);
                atomicAdd(base + 1, y);
                atomicAdd(base + 2, z);
                atomicAdd(base + 3, 1.0f);
            }
        }
    }
    __syncthreads();
    for (int i = threadIdx.x; i < MAXEV * KMAX * 4; i += blockDim.x) {
        float v = ls[i];
        if (v != 0.0f) atomicAdd(&ws[STAT_OFF + i], v);
    }
}

// Per event: K, packed av4, invRep, invDist, and pre-swizzled WMMA A-fragments.
__global__ void ll_stats_kernel(const long long* __restrict__ rs, float* ws, int nEvents) {
    int e = blockIdx.x;
    if (e >= nEvents || e >= MAXEV) return;
    const float4* stat = (const float4*)(ws + STAT_OFF) + e * KMAX;
    float4* av4 = (float4*)(ws + AV4_OFF) + e * KPAD;
    float* invRep = ws + IR_OFF + e * KPAD;
    float* invDist = ws + ID_OFF + e * KPAD;
    float2* afrag = (float2*)(ws + AF_OFF) + e * (KTILES * 32);
    long long a = rs[e], b = rs[e + 1];
    float nTot = (float)(b - a);
    bool valid = (b - a) >= 20;
    int k = threadIdx.x;
    __shared__ float red[256];
    float cnt = 0.0f, ex = 0.0f;
    if (k < KMAX) {
        cnt = stat[k].w;
        ex = (valid && cnt > 0.0f) ? 1.0f : 0.0f;
    }
    red[k] = ex;
    __syncthreads();
    for (int s = 128; s > 0; s >>= 1) {
        if (k < s) red[k] += red[k + s];
        __syncthreads();
    }
    float K = red[0];
    if (k < KPAD) {
        float4 o = make_float4(0.0f, 0.0f, 0.0f, 0.0f);
        float ir = 0.0f, id = 0.0f;
        if (k < KMAX && ex > 0.0f) {
            float4 s4 = stat[k];
            float inv = 1.0f / cnt;
            float ax = s4.x * inv, ay = s4.y * inv, az = s4.z * inv;
            o = make_float4(-2.0f * ax, -2.0f * ay, -2.0f * az,
                            ax * ax + ay * ay + az * az);
            id = 1.0f / (cnt * K);
            ir = 1.0f / ((nTot - cnt) * (K + 0.001f));
        }
        av4[k] = o;
        invRep[k] = ir;
        invDist[k] = id;
        // Pre-swizzled A fragment: 16x4 (M x K) f32 WMMA layout.
        // Lanes 0-15 (M=k%16) hold K={0,1} = (-2ax,-2ay); lanes 16-31 hold K={2,3} = (-2az,|a|^2).
        int kt = k >> 4, li = k & 15;
        afrag[kt * 32 + li]      = make_float2(o.x, o.y);
        afrag[kt * 32 + 16 + li] = make_float2(o.z, o.w);
    }
}

// Main loss kernel: each wave owns a 16-point tile; nkt (=13) WMMA cluster tiles.
// nkt is a runtime argument so the K-tile loop CANNOT be fully unrolled
// (full unroll keeps 13x8 accumulator VGPRs live and spills to scratch).
__global__ void ll_main_kernel(const float* __restrict__ coords,
                               const int* __restrict__ tidx,
                               const long long* __restrict__ rs,
                               const float* __restrict__ ws,
                               float* out, int nPoints, int nEvents, int nkt) {
    const int lane = threadIdx.x & 31;
    const int wave = threadIdx.x >> 5;
    const bool hi = lane >= 16;
    const int li = lane & 15;
    const int tile = blockIdx.x * 8 + wave;
    const int totalTiles = (nPoints + 15) >> 4;

    float contrib = 0.0f;
    if (tile < totalTiles) {
        int p0 = tile * 16;
        int e = 0;
        for (int i = 0; i < nEvents; ++i)
            if ((long long)p0 >= rs[i]) e = i;
        long long a = rs[e], b = rs[e + 1];
        bool evValid = ((b - a) >= 20) && (e < MAXEV);
        int p = p0 + li;
        int pc = p < nPoints ? p : nPoints - 1;
        bool pValid = evValid && (p < nPoints) && ((long long)p >= a) && ((long long)p < b);

        float x = coords[3 * pc + 0];
        float y = coords[3 * pc + 1];
        float z = coords[3 * pc + 2];
        int t = tidx[pc];
        float c2 = x * x + y * y + z * z;

        int ec = e < MAXEV ? e : 0;
        const float4* av4 = (const float4*)(ws + AV4_OFF) + ec * KPAD;
        const float* invRep = ws + IR_OFF + ec * KPAD;
        const float* invDist = ws + ID_OFF + ec * KPAD;
        const float2* afrag = (const float2*)(ws + AF_OFF) + ec * (KTILES * 32);

        // B fragment: 4x16 (K x N).  Lanes 0-15 supply K={0,1}=(x,y); lanes 16-31 K={2,3}=(z,1).
        v2f B;
        B.x = hi ? z : x;
        B.y = hi ? 1.0f : y;

        // C seed: every accumulator VGPR of a lane maps to its own point (N=lane%16),
        // so broadcasting c2 yields d2 = |c|^2 - 2 av.c + |av|^2 directly from the WMMA.
        v8f cseed;
#pragma unroll
        for (int v = 0; v < 8; ++v) cseed[v] = c2;

        // Two packed accumulator chains -> v_pk_fma_f32 with ILP.
        v2f acc0 = {0.0f, 0.0f};
        v2f acc1 = {0.0f, 0.0f};
        for (int kt = 0; kt < nkt; ++kt) {
            int kbase = kt * 16 + (hi ? 8 : 0);
            float cv[8];
#if USE_WMMA
            // A fragment: pre-swizzled coalesced b64 load, no per-lane selects.
            float2 af = afrag[kt * 32 + lane];
            v2f A;
            A.x = af.x;
            A.y = af.y;
            v8f d = __builtin_amdgcn_wmma_f32_16x16x4_f32(
                false, A, false, B, (short)0, cseed, false, false);
#pragma unroll
            for (int v = 0; v < 8; ++v) cv[v] = d[v];
#else
#pragma unroll
            for (int v = 0; v < 8; ++v) {
                float4 av = av4[kbase + v];
                cv[v] = c2 + fmaf(av.x, x, fmaf(av.y, y, fmaf(av.z, z, av.w)));
            }
#endif
            float4 r0 = *(const float4*)(invRep + kbase);
            float4 r1 = *(const float4*)(invRep + kbase + 4);
            float rr[8] = {r0.x, r0.y, r0.z, r0.w, r1.x, r1.y, r1.z, r1.w};
            // exp(-max(d,0)) == min(exp(-d),1): clamp01 folds into v_exp's clamp
            // output modifier, so no separate max/canonicalize instructions.
#pragma unroll
            for (int v = 0; v < 8; v += 2) {
                v2f ev;
                ev.x = clamp01(__expf(-cv[v]));
                ev.y = clamp01(__expf(-cv[v + 1]));
                v2f rp;
                rp.x = rr[v];
                rp.y = rr[v + 1];
                if ((v >> 1) & 1) acc1 += ev * rp;   // packed fma chain 1
                else             acc0 += ev * rp;    // packed fma chain 0
            }
        }
        float repAcc = (acc0.x + acc1.x) + (acc0.y + acc1.y);

        float noise_w = (t < 0) ? 0.1f : 1.0f;
        contrib = noise_w * repAcc;
        // Own-cluster correction + distance loss: apply once per point (lo half only).
        if (!hi && t >= 0 && t < KMAX) {
            float4 a4 = av4[t];
            float d2o = clamp_pos(c2 + fmaf(a4.x, x, fmaf(a4.y, y, fmaf(a4.z, z, a4.w))));
            contrib -= noise_w * __expf(-d2o) * invRep[t];
            contrib += __logf(2.718281828f * d2o + 1.0f) * invDist[t];
        }
        if (!pValid) contrib = 0.0f;
    }

    // wave32 reduction, then block reduction, one atomic per block
#pragma unroll
    for (int m = 16; m >= 1; m >>= 1) contrib += __shfl_xor(contrib, m, 32);
    __shared__ float wsum[8];
    if (lane == 0) wsum[wave] = contrib;
    __syncthreads();
    if (threadIdx.x == 0) {
        float s = 0.0f;
#pragma unroll
        for (int w = 0; w < 8; ++w) s += wsum[w];
        atomicAdd(out, s);
    }
}

extern "C" void kernel_launch(void* const* d_in, const int* in_sizes, int n_in,
                              void* d_out, int out_size, void* d_ws, size_t ws_size,
                              hipStream_t stream) {
    const float* coords = (const float*)d_in[0];
    const int* tidx = (const int*)d_in[1];
    const long long* rs = (const long long*)d_in[2];
    float* out = (float*)d_out;
    float* ws = (float*)d_ws;

    int nPoints = in_sizes[0] / 3;
    int nEvents = in_sizes[2] - 1;
    if (nEvents > MAXEV) nEvents = MAXEV;

    ll_zero_kernel<<<(STAT_SIZE + 255) / 256, 256, 0, stream>>>(ws, out);
    ll_accum_kernel<<<64, 256, 0, stream>>>(coords, tidx, rs, ws, nPoints, nEvents);
    ll_stats_kernel<<<nEvents, 256, 0, stream>>>(rs, ws, nEvents);

    int totalTiles = (nPoints + 15) / 16;
    int blocks = (totalTiles + 7) / 8;
    ll_main_kernel<<<blocks, 256, 0, stream>>>(coords, tidx, rs, ws, out,
                                               nPoints, nEvents, KTILES);
}